// Ricker_50148038148511
// MI455X (gfx1250) — compile-verified
//
#include <hip/hip_runtime.h>

// Ricker-map nonlinear scan: n[i+1] = n[i] * exp(alpha*(1 - beta*n[i] + bx*t + cx*t^2))
// Inherently sequential (state enters the exponent nonlinearly) -> latency-bound.
//
// Change of variable m = G*n with G = alpha*beta*log2(e):
//   m' = m * E(t) * exp2(-m),   E(t) = exp2(A*(1 + bx*t + cx*t^2)), A = alpha*log2(e)
// The negate on exp2's operand is a free VALU source modifier, so the
// per-step dependent chain is exactly:  v_exp_f32(-m) -> v_mul   (m*E overlaps).
// Output n = m * (1/G) is off-chain (feeds only the store).
//
// Single wave32: lane 0 runs the chain; all 32 lanes stage Temp chunks through
// a double-buffered LDS pipeline and apply the E(t) transform wave-parallel.

#define CHUNK 2048                      // floats per pipeline chunk (8 KB)
#define LANES 32
#define V4PL  (CHUNK / (LANES * 4))     // float4s per lane per chunk = 16

__launch_bounds__(LANES, 1)
__global__ void ricker_scan_kernel(const float* __restrict__ N0p,
                                   const float* __restrict__ Temp,
                                   const float* __restrict__ params,
                                   float* __restrict__ out,
                                   int T) {
    __shared__ float buf[2][CHUNK];     // double buffer, 16 KB of 320 KB LDS

    const int lane = threadIdx.x;

    const float alpha = params[0];
    const float beta  = params[1];
    const float bx    = params[2];
    const float cx    = params[3];
    const float LOG2E = 1.44269504088896340736f;
    const float A    = alpha * LOG2E;          // env-term exp2 scale
    const float G    = alpha * beta * LOG2E;   // density-term exp2 scale
    const float invG = 1.0f / G;

    const float n0 = *N0p;
    if (lane == 0) out[0] = n0;

    const int steps = T - 1;                   // recurrence uses Temp[0..steps-1]
    if (steps <= 0) return;

    const int nchunks = (steps + CHUNK - 1) / CHUNK;
    const int maxv4   = (T >> 2) - 1;          // clamp for float4 reads of Temp
    const float4* t4  = reinterpret_cast<const float4*>(Temp);

    float m = G * n0;                          // scaled state
    float4 reg[V4PL];

    // ---- stage chunk 0: load, transform to E(t), park in LDS ----
    #pragma unroll
    for (int j = 0; j < V4PL; ++j) {
        int v = j * LANES + lane;
        reg[j] = t4[min(v, maxv4)];
    }
    #pragma unroll
    for (int j = 0; j < V4PL; ++j) {
        float4 t = reg[j], e;
        e.x = exp2f(A * fmaf(t.x, fmaf(cx, t.x, bx), 1.0f));
        e.y = exp2f(A * fmaf(t.y, fmaf(cx, t.y, bx), 1.0f));
        e.z = exp2f(A * fmaf(t.z, fmaf(cx, t.z, bx), 1.0f));
        e.w = exp2f(A * fmaf(t.w, fmaf(cx, t.w, bx), 1.0f));
        reinterpret_cast<float4*>(&buf[0][0])[j * LANES + lane] = e;
    }
    __syncthreads();   // s_wait_dscnt (single-wave WG: barrier is a NOP)

    for (int c = 0; c < nchunks; ++c) {
        const int cur = c & 1;

        // Issue global loads for chunk c+1 now; they fly while lane 0 computes.
        if (c + 1 < nchunks) {
            #pragma unroll
            for (int j = 0; j < V4PL; ++j) {
                int v = (c + 1) * (CHUNK / 4) + j * LANES + lane;
                reg[j] = t4[min(v, maxv4)];
            }
            // gfx1250 global_prefetch_b8: pull chunk c+2 toward L2 (8 MB << 192 MB)
            if (c + 2 < nchunks) {
                const char* p = reinterpret_cast<const char*>(Temp)
                              + (size_t)(c + 2) * CHUNK * sizeof(float);
                __builtin_prefetch(p + lane * 256, 0, 1);
            }
        }

        // ---- serial recurrence over chunk c (lane 0 only) ----
        // Dependent chain per step: v_exp_f32(-m) -> v_mul.  (m*E and the
        // output scale m*invG are independent of the chain tail.)
        const int base = c * CHUNK;
        const int cnt  = min(CHUNK, steps - base);
        if (lane == 0) {
            #pragma unroll 8
            for (int i = 0; i < cnt; ++i) {
                float e  = buf[cur][i];        // ds_load, address independent of m
                float ex = exp2f(-m);          // v_exp_f32 with neg src modifier
                float p  = m * e;              // overlaps the exp
                m = p * ex;                    // chain tail
                out[base + i + 1] = m * invG;  // off-chain, fire-and-forget store
            }
        }
        __syncthreads();

        // ---- transform + park chunk c+1 into the other LDS buffer ----
        if (c + 1 < nchunks) {
            #pragma unroll
            for (int j = 0; j < V4PL; ++j) {
                float4 t = reg[j], e;
                e.x = exp2f(A * fmaf(t.x, fmaf(cx, t.x, bx), 1.0f));
                e.y = exp2f(A * fmaf(t.y, fmaf(cx, t.y, bx), 1.0f));
                e.z = exp2f(A * fmaf(t.z, fmaf(cx, t.z, bx), 1.0f));
                e.w = exp2f(A * fmaf(t.w, fmaf(cx, t.w, bx), 1.0f));
                reinterpret_cast<float4*>(&buf[cur ^ 1][0])[j * LANES + lane] = e;
            }
        }
        __syncthreads();
    }
}

extern "C" void kernel_launch(void* const* d_in, const int* in_sizes, int n_in,
                              void* d_out, int out_size, void* d_ws, size_t ws_size,
                              hipStream_t stream) {
    (void)n_in; (void)d_ws; (void)ws_size; (void)out_size;
    const float* N0     = (const float*)d_in[0];   // scalar (size 1)
    const float* Temp   = (const float*)d_in[1];   // [T]
    const float* params = (const float*)d_in[2];   // [alpha, beta, bx, cx]
    float*       out    = (float*)d_out;           // [T]
    const int T = in_sizes[1];

    // Single wave32: the scan is a serial dependency chain; more blocks can't help.
    ricker_scan_kernel<<<1, LANES, 0, stream>>>(N0, Temp, params, out, T);
}